// PointPillarScatter_lane_58935541236024
// MI455X (gfx1250) — compile-verified
//
#include <hip/hip_runtime.h>
#include <stddef.h>

// ---------------------------------------------------------------------------
// PointPillarScatter for MI455X (gfx1250).
// Pure memory-movement op: zero a [8,64,496*432] f32 canvas (~439 MB), then
// scatter-transpose 200k pillars x 64 channels (~51 MB) into it.
// ~545 MB total -> ~23 us at 23.3 TB/s. No matmul => no WMMA; we instead use
// the CDNA5 async global->LDS copy path (ASYNCcnt) + LDS transpose + NT fill.
// ---------------------------------------------------------------------------

namespace {
constexpr int kNX = 432;
constexpr int kNY = 496;
constexpr int kC  = 64;
constexpr int kS  = kNX * kNY;   // 214272 spatial cells per batch
constexpr int kTP = 32;          // pillars per block tile
constexpr int kThreads = 256;    // 8 wave32 per block
}

typedef __attribute__((ext_vector_type(4))) float v4f;
typedef __attribute__((address_space(1))) int g_i32;   // global (AS1)
typedef __attribute__((address_space(3))) int l_i32;   // LDS (AS3)

#ifndef __has_builtin
#define __has_builtin(x) 0
#endif

#if __has_builtin(__builtin_amdgcn_global_load_async_to_lds_b32)
#define PPS_ASYNC_LDS 1
#else
#define PPS_ASYNC_LDS 0
#endif

// ---------------------------------------------------------------------------
// Kernel 1: zero the canvas with 16B non-temporal stores (canvas is 439 MB,
// >> 192 MB L2, so NT avoids polluting L2 ahead of the scatter phase).
// ---------------------------------------------------------------------------
__global__ __launch_bounds__(kThreads)
void pps_zero_kernel(v4f* __restrict__ out, long long n4) {
  long long i = (long long)blockIdx.x * blockDim.x + threadIdx.x;
  if (i < n4) {
    v4f z = {0.f, 0.f, 0.f, 0.f};
    __builtin_nontemporal_store(z, &out[i]);
  }
}

// ---------------------------------------------------------------------------
// Kernel 2: scatter with LDS transpose.
//   Phase 1: stage 32 pillars x 64 ch (8 KB) into LDS, fully coalesced,
//            via CDNA5 async global->LDS copies when available.
//   Phase 2: wave w (of 8) owns channels [8w, 8w+8); lane = local pillar.
//            LDS row pitch 65 floats -> bank (pl+c)%64, conflict-free reads.
//            32 lanes write 32 consecutive spatial cells per channel
//            (coords are contiguous per batch here) -> coalesced 128B stores.
// ---------------------------------------------------------------------------
__global__ __launch_bounds__(kThreads)
void pps_scatter_kernel(const float* __restrict__ feat,
                        const int*   __restrict__ coords,
                        float*       __restrict__ out,
                        int P) {
  __shared__ float tile[kTP][kC + 1];   // +1 pad: conflict-free transpose reads

  const int p0 = blockIdx.x * kTP;

  // ---- Phase 1: stage tile into LDS (2048 floats, 8 per thread) ----------
#if PPS_ASYNC_LDS
  #pragma unroll
  for (int k = 0; k < 8; ++k) {
    const int i  = threadIdx.x + k * kThreads;  // 0..2047, contiguous in feat
    const int pl = i >> 6;
    const int c  = i & 63;
    const int p  = p0 + pl;
    if (p < P) {
      void* gp = (void*)(feat + (size_t)p * kC + c);
      void* lp = (void*)(&tile[pl][c]);
      __builtin_amdgcn_global_load_async_to_lds_b32(
          (g_i32*)gp, (l_i32*)lp, /*offset=*/0, /*cpol=*/0);
    } else {
      tile[pl][c] = 0.f;
    }
  }
#if __has_builtin(__builtin_amdgcn_s_wait_asynccnt)
  __builtin_amdgcn_s_wait_asynccnt(0);
#else
  asm volatile("s_wait_asynccnt 0" ::: "memory");
#endif
#else
  #pragma unroll
  for (int k = 0; k < 8; ++k) {
    const int i  = threadIdx.x + k * kThreads;
    const int pl = i >> 6;
    const int c  = i & 63;
    const int p  = p0 + pl;
    tile[pl][c] = (p < P) ? feat[(size_t)p * kC + c] : 0.f;
  }
#endif
  __syncthreads();

  // ---- Phase 2: transposed, coalesced scatter ----------------------------
  const int pl = threadIdx.x & (kTP - 1);   // lane -> local pillar
  const int cg = threadIdx.x >> 5;          // wave -> channel group (0..7)
  const int p  = p0 + pl;
  if (p < P) {
    const int b = coords[4 * p + 0];
    const int z = coords[4 * p + 1];
    const int y = coords[4 * p + 2];
    const int x = coords[4 * p + 3];
    const size_t sidx = (size_t)z + (size_t)y * kNX + (size_t)x;
    const size_t base = (size_t)b * kC * kS + sidx;
    #pragma unroll
    for (int cc = 0; cc < kC / 8; ++cc) {
      const int c = cg * 8 + cc;
      out[base + (size_t)c * kS] = tile[pl][c];
    }
  }
}

// ---------------------------------------------------------------------------
// Launcher: d_in[0] = pillar_features f32 [P*64], d_in[1] = coords i32 [P*4],
// d_out = f32 [8*64*496*432]. Both kernels on `stream` (graph-capture safe).
// ---------------------------------------------------------------------------
extern "C" void kernel_launch(void* const* d_in, const int* in_sizes, int n_in,
                              void* d_out, int out_size, void* d_ws, size_t ws_size,
                              hipStream_t stream) {
  (void)n_in; (void)d_ws; (void)ws_size;

  const float* feat   = (const float*)d_in[0];
  const int*   coords = (const int*)d_in[1];
  float*       out    = (float*)d_out;

  const int P = in_sizes[0] / kC;                 // 200000 pillars

  // Zero fill: out_size = 109,707,264 floats (divisible by 4).
  const long long n4 = (long long)out_size / 4;
  const long long zb = (n4 + kThreads - 1) / kThreads;
  pps_zero_kernel<<<(unsigned)zb, kThreads, 0, stream>>>((v4f*)out, n4);

  // Scatter: one 32-pillar tile per block.
  const int sb = (P + kTP - 1) / kTP;             // 6250 blocks
  pps_scatter_kernel<<<sb, kThreads, 0, stream>>>(feat, coords, out, P);
}